// RNNDecoder_14121852469434
// MI455X (gfx1250) — compile-verified
//
#include <hip/hip_runtime.h>
#include <hip/hip_bf16.h>

// Problem constants (match reference exactly)
#define BB   128
#define SS   400
#define EE   512
#define HH   512
#define MMD  512
#define VV   50000
#define OOVN 64
#define CCN  10
#define VO   (VV + OOVN)

typedef float v2f __attribute__((ext_vector_type(2)));
typedef float v8f __attribute__((ext_vector_type(8)));

__device__ __forceinline__ float sigmoidf_(float x) { return 1.0f / (1.0f + expf(-x)); }

// ---------------------------------------------------------------------------
// Generic fp32 WMMA GEMM:  C[M,N] = A[M,K] @ W[N,K]^T + bias[N]
// grid.x = N/16 ; block = 256 threads = 8 waves ; wave w -> rows [16w,16w+16)
// Uses V_WMMA_F32_16X16X4_F32 (full fp32 precision; workload is HBM-bound,
// so fp32 matrix pipe is more than sufficient and avoids conversion traffic).
//
// f32 16x16x4 fragment layout (ISA 7.12.2):
//   A: lane l (0-15) holds A[M=l][K=k+0..1], lane l+16 holds A[M=l][K=k+2..3]
//   B: lane n (0-15) holds B[K=k+0..1][n],  lane n+16 holds B[K=k+2..3][n]
//   Since B[k][n] = W[n][k] and W is row-major in K, both frags are float2 loads.
//   C: VGPR r, lanes 0-15 -> C[M=r][N=lane]; lanes 16-31 -> C[M=8+r][N=lane-16]
// ---------------------------------------------------------------------------
__global__ __launch_bounds__(256)
void wmma_gemm_nt(const float* __restrict__ A, const float* __restrict__ W,
                  const float* __restrict__ bias, float* __restrict__ C,
                  int M, int N, int K) {
  const int wave = threadIdx.x >> 5;
  const int lane = threadIdx.x & 31;
  const int hf   = lane >> 4;      // which K-pair half
  const int l15  = lane & 15;
  const int n0   = blockIdx.x << 4;
  const int m0   = wave << 4;
  if (m0 >= M) return;             // wave-uniform, EXEC stays all-ones

  const float* Ap = A + (size_t)(m0 + l15) * K + (hf << 1);
  const float* Bp = W + (size_t)(n0 + l15) * K + (hf << 1);

  v8f acc = {};
#pragma unroll 4
  for (int k = 0; k < K; k += 4) {
    v2f a = *(const v2f*)(Ap + k);
    v2f b = *(const v2f*)(Bp + k);
    acc = __builtin_amdgcn_wmma_f32_16x16x4_f32(
        /*neg_a=*/false, a, /*neg_b=*/false, b,
        /*c_mod=*/(short)0, acc, /*reuse_a=*/false, /*reuse_b=*/false);
  }

  const float bn = bias ? bias[n0 + l15] : 0.0f;
  const int mrow = m0 + (hf << 3);
#pragma unroll
  for (int r = 0; r < 8; ++r)
    C[(size_t)(mrow + r) * N + n0 + l15] = acc[r] + bn;
}

// ---------------------------------------------------------------------------
// Embedding gather: y_emb[b,:] = embedding[y[b],:]
// ---------------------------------------------------------------------------
__global__ void k_embed(const int* __restrict__ y, const float* __restrict__ emb,
                        float* __restrict__ y_emb) {
  const int b = blockIdx.x;
  const int row = y[b];
  for (int e = threadIdx.x; e < EE; e += blockDim.x)
    y_emb[b * EE + e] = emb[(size_t)row * EE + e];
}

// ---------------------------------------------------------------------------
// GRU gate nonlinearity: h_next = (1-z)*n + z*h0
// ---------------------------------------------------------------------------
__global__ void k_gru_gates(const float* __restrict__ gi, const float* __restrict__ gh,
                            const float* __restrict__ h0, float* __restrict__ h_next) {
  const int b = blockIdx.x;
  const size_t base = (size_t)b * 3 * HH;
  for (int j = threadIdx.x; j < HH; j += blockDim.x) {
    float ir = gi[base + j],          hr = gh[base + j];
    float iz = gi[base + HH + j],     hz = gh[base + HH + j];
    float in_ = gi[base + 2*HH + j],  hn = gh[base + 2*HH + j];
    float r = sigmoidf_(ir + hr);
    float z = sigmoidf_(iz + hz);
    float n = tanhf(in_ + r * hn);
    h_next[b * HH + j] = (1.0f - z) * n + z * h0[b * HH + j];
  }
}

// ---------------------------------------------------------------------------
// scores[b,s] = dot(memory_bank[b,s,:], proj[b,:])  -- one wave per dot
// ---------------------------------------------------------------------------
__global__ __launch_bounds__(256)
void k_scores(const float* __restrict__ mb, const float* __restrict__ proj,
              float* __restrict__ scores) {
  const int gw = (int)((blockIdx.x * 256 + threadIdx.x) >> 5);
  const int lane = threadIdx.x & 31;
  if (gw >= BB * SS) return;
  const int b = gw / SS, s = gw - b * SS;
  const float4* m4 = (const float4*)(mb + ((size_t)b * SS + s) * MMD);
  const float4* p4 = (const float4*)(proj + (size_t)b * MMD);
  float acc = 0.0f;
  for (int i = lane; i < MMD / 4; i += 32) {
    float4 mv = m4[i], pv = p4[i];
    acc += mv.x * pv.x + mv.y * pv.y + mv.z * pv.z + mv.w * pv.w;
  }
#pragma unroll
  for (int off = 16; off > 0; off >>= 1) acc += __shfl_xor(acc, off);
  if (lane == 0) scores[(size_t)b * SS + s] = acc;
}

// ---------------------------------------------------------------------------
// Masked softmax: attn = exp(s - max)*mask / sum(exp(s - max)*mask)
// (algebraically identical to softmax->mask->renormalize in the reference)
// ---------------------------------------------------------------------------
__global__ __launch_bounds__(256)
void k_masked_softmax(const float* __restrict__ scores, const float* __restrict__ mask,
                      float* __restrict__ attn) {
  const int b = blockIdx.x;
  __shared__ float red[8];
  __shared__ float bval;
  const int tid = threadIdx.x, lane = tid & 31, wv = tid >> 5;

  float mx = -3.4e38f;
  for (int s = tid; s < SS; s += 256) mx = fmaxf(mx, scores[b * SS + s]);
#pragma unroll
  for (int off = 16; off > 0; off >>= 1) mx = fmaxf(mx, __shfl_xor(mx, off));
  if (lane == 0) red[wv] = mx;
  __syncthreads();
  if (tid == 0) { float m = red[0]; for (int i = 1; i < 8; ++i) m = fmaxf(m, red[i]); bval = m; }
  __syncthreads();
  const float rowmax = bval;

  float sum = 0.0f;
  for (int s = tid; s < SS; s += 256)
    sum += expf(scores[b * SS + s] - rowmax) * mask[b * SS + s];
#pragma unroll
  for (int off = 16; off > 0; off >>= 1) sum += __shfl_xor(sum, off);
  __syncthreads();
  if (lane == 0) red[wv] = sum;
  __syncthreads();
  if (tid == 0) { float t = 0; for (int i = 0; i < 8; ++i) t += red[i]; bval = t; }
  __syncthreads();
  const float inv = 1.0f / bval;

  for (int s = tid; s < SS; s += 256)
    attn[b * SS + s] = expf(scores[b * SS + s] - rowmax) * mask[b * SS + s] * inv;
}

// ---------------------------------------------------------------------------
// context[b,m] = sum_s attn[b,s] * memory_bank[b,s,m]   (attn staged in LDS)
// ---------------------------------------------------------------------------
__global__ __launch_bounds__(512)
void k_context(const float* __restrict__ mb, const float* __restrict__ attn,
               float* __restrict__ ctx) {
  __shared__ float a[SS];
  const int b = blockIdx.x;
  for (int s = threadIdx.x; s < SS; s += blockDim.x) a[s] = attn[b * SS + s];
  __syncthreads();
  const int m = threadIdx.x;  // blockDim == MMD == 512
  const float* base = mb + (size_t)b * SS * MMD + m;
  float acc = 0.0f;
  for (int s = 0; s < SS; ++s) acc += a[s] * base[(size_t)s * MMD];
  ctx[b * MMD + m] = acc;
}

// ---------------------------------------------------------------------------
// vocab_in = concat(context, h_last)
// ---------------------------------------------------------------------------
__global__ void k_pack(const float* __restrict__ ctx, const float* __restrict__ h,
                       float* __restrict__ vocab_in) {
  const int b = blockIdx.x;
  for (int j = threadIdx.x; j < HH + MMD; j += blockDim.x)
    vocab_in[(size_t)b * (HH + MMD) + j] = (j < MMD) ? ctx[b * MMD + j]
                                                     : h[b * HH + (j - MMD)];
}

// ---------------------------------------------------------------------------
// p_gen[b] = sigmoid([ctx|h|y_emb] . Wp + bp)  -- one wave per batch row
// ---------------------------------------------------------------------------
__global__ __launch_bounds__(256)
void k_pgen(const float* __restrict__ ctx, const float* __restrict__ h,
            const float* __restrict__ yemb, const float* __restrict__ Wp,
            const float* __restrict__ bp, float* __restrict__ pgen) {
  const int b = (int)((blockIdx.x * 256 + threadIdx.x) >> 5);
  const int lane = threadIdx.x & 31;
  if (b >= BB) return;
  float acc = 0.0f;
  for (int i = lane; i < MMD; i += 32) acc += ctx[b * MMD + i] * Wp[i];
  for (int i = lane; i < HH;  i += 32) acc += h[b * HH + i] * Wp[MMD + i];
  for (int i = lane; i < EE;  i += 32) acc += yemb[b * EE + i] * Wp[MMD + HH + i];
#pragma unroll
  for (int off = 16; off > 0; off >>= 1) acc += __shfl_xor(acc, off);
  if (lane == 0) pgen[b] = sigmoidf_(acc + bp[0]);
}

// ---------------------------------------------------------------------------
// Per-row max and sum(exp(l-max)) over V=50000 logits
// ---------------------------------------------------------------------------
__global__ __launch_bounds__(256)
void k_rowstats(const float* __restrict__ logits, float* __restrict__ mx,
                float* __restrict__ sm) {
  const int b = blockIdx.x;
  __shared__ float red[8];
  __shared__ float bval;
  const int tid = threadIdx.x, lane = tid & 31, wv = tid >> 5;
  const float* row = logits + (size_t)b * VV;

  float m = -3.4e38f;
  for (int v = tid; v < VV; v += 256) m = fmaxf(m, row[v]);
#pragma unroll
  for (int off = 16; off > 0; off >>= 1) m = fmaxf(m, __shfl_xor(m, off));
  if (lane == 0) red[wv] = m;
  __syncthreads();
  if (tid == 0) { float t = red[0]; for (int i = 1; i < 8; ++i) t = fmaxf(t, red[i]); bval = t; }
  __syncthreads();
  const float rm = bval;

  float s = 0.0f;
  for (int v = tid; v < VV; v += 256) s += expf(row[v] - rm);
#pragma unroll
  for (int off = 16; off > 0; off >>= 1) s += __shfl_xor(s, off);
  __syncthreads();
  if (lane == 0) red[wv] = s;
  __syncthreads();
  if (tid == 0) { float t = 0; for (int i = 0; i < 8; ++i) t += red[i]; mx[b] = rm; sm[b] = t; }
}

// ---------------------------------------------------------------------------
// final[b,v] = p_gen*softmax(logits)[b,v] for v<V, 0 for OOV tail
// Fully overwrites the region every call (harness poisons d_out).
// ---------------------------------------------------------------------------
__global__ __launch_bounds__(256)
void k_final(const float* __restrict__ logits, const float* __restrict__ mx,
             const float* __restrict__ sm, const float* __restrict__ pgen,
             float* __restrict__ fin) {
  const int b = blockIdx.y;
  const int v = blockIdx.x * 256 + threadIdx.x;
  if (v >= VO) return;
  float out = 0.0f;
  if (v < VV) out = pgen[b] * expf(logits[(size_t)b * VV + v] - mx[b]) / sm[b];
  fin[(size_t)b * VO + v] = out;
}

// ---------------------------------------------------------------------------
// Scatter-adds (indices may repeat within a row -> global float atomics)
// ---------------------------------------------------------------------------
__global__ void k_scatter_attn(const int* __restrict__ src_oov,
                               const float* __restrict__ attn,
                               const float* __restrict__ pgen,
                               float* __restrict__ fin) {
  const int t = blockIdx.x * 256 + threadIdx.x;
  if (t >= BB * SS) return;
  const int b = t / SS;
  atomicAdd(&fin[(size_t)b * VO + src_oov[t]], (1.0f - pgen[b]) * attn[t]);
}

__global__ void k_scatter_cls(const int* __restrict__ cls_oov,
                              const float* __restrict__ cls_pred,
                              const float* __restrict__ pgen,
                              float* __restrict__ fin) {
  const int t = blockIdx.x * 256 + threadIdx.x;
  if (t >= BB * CCN) return;
  const int b = t / CCN;
  atomicAdd(&fin[(size_t)b * VO + cls_oov[t]], (1.0f - pgen[b]) * cls_pred[t]);
}

// ---------------------------------------------------------------------------
extern "C" void kernel_launch(void* const* d_in, const int* in_sizes, int n_in,
                              void* d_out, int out_size, void* d_ws, size_t ws_size,
                              hipStream_t stream) {
  const int*   y        = (const int*)  d_in[0];
  const float* h0       = (const float*)d_in[1];   // [1,B,H]
  const float* mem_bank = (const float*)d_in[2];   // [B,S,M]
  const float* src_mask = (const float*)d_in[3];   // [B,S]
  // d_in[4] = max_num_oovs scalar (compile-time 64 here)
  const int*   src_oov  = (const int*)  d_in[5];   // [B,S]
  const float* cls_pred = (const float*)d_in[6];   // [B,C]
  const int*   cls_oov  = (const int*)  d_in[7];   // [B,C]
  const float* embedding= (const float*)d_in[8];   // [V,E]
  const float* W_ih     = (const float*)d_in[9];   // [3H,E]
  const float* W_hh     = (const float*)d_in[10];  // [3H,H]
  const float* b_ih     = (const float*)d_in[11];
  const float* b_hh     = (const float*)d_in[12];
  const float* attn_W   = (const float*)d_in[13];  // [M,H]
  const float* W1       = (const float*)d_in[14];  // [H,H+M]
  const float* b1       = (const float*)d_in[15];
  const float* W2       = (const float*)d_in[16];  // [V,H]
  const float* b2       = (const float*)d_in[17];
  const float* Wp       = (const float*)d_in[18];  // [1,E+H+M]
  const float* bp       = (const float*)d_in[19];

  // Output regions (return order: final_dist, h_next, context, attn_dist, p_gen)
  float* fin    = (float*)d_out;                    // [B, V+OOV]
  float* h_next = fin + (size_t)BB * VO;            // [B, H] (== h_last)
  float* ctx    = h_next + (size_t)BB * HH;         // [B, M]
  float* attn   = ctx + (size_t)BB * MMD;           // [B, S]
  float* pgen   = attn + (size_t)BB * SS;           // [B]

  // Workspace (~28.7 MB of fp32)
  float* ws       = (float*)d_ws;
  float* y_emb    = ws;                             // B*E
  float* gi       = y_emb + (size_t)BB * EE;        // B*3H
  float* gh       = gi + (size_t)BB * 3 * HH;       // B*3H
  float* proj     = gh + (size_t)BB * 3 * HH;       // B*M
  float* scores   = proj + (size_t)BB * MMD;        // B*S
  float* vocab_in = scores + (size_t)BB * SS;       // B*(H+M)
  float* mx       = vocab_in + (size_t)BB * (HH + MMD); // B
  float* sm       = mx + BB;                        // B
  float* logits   = sm + BB;                        // B*V
  float* hid      = logits + (size_t)BB * VV;       // B*H

  // 1. embedding gather
  k_embed<<<BB, 256, 0, stream>>>(y, embedding, y_emb);
  // 2. GRU input/hidden GEMMs (WMMA fp32)
  wmma_gemm_nt<<<(3 * HH) / 16, 256, 0, stream>>>(y_emb, W_ih, b_ih, gi, BB, 3 * HH, EE);
  wmma_gemm_nt<<<(3 * HH) / 16, 256, 0, stream>>>(h0,    W_hh, b_hh, gh, BB, 3 * HH, HH);
  // 3. gates -> h_last (written straight to d_out h_next region)
  k_gru_gates<<<BB, 256, 0, stream>>>(gi, gh, h0, h_next);
  // 4. attention projection (WMMA)
  wmma_gemm_nt<<<MMD / 16, 256, 0, stream>>>(h_next, attn_W, nullptr, proj, BB, MMD, HH);
  // 5. attention scores + masked softmax + context
  k_scores<<<(BB * SS * 32) / 256, 256, 0, stream>>>(mem_bank, proj, scores);
  k_masked_softmax<<<BB, 256, 0, stream>>>(scores, src_mask, attn);
  k_context<<<BB, 512, 0, stream>>>(mem_bank, attn, ctx);
  // 6. vocab path
  k_pack<<<BB, 256, 0, stream>>>(ctx, h_next, vocab_in);
  wmma_gemm_nt<<<HH / 16, 256, 0, stream>>>(vocab_in, W1, b1, hid, BB, HH, HH + MMD);
  // 7. p_gen
  k_pgen<<<(BB * 32 + 255) / 256, 256, 0, stream>>>(ctx, h_next, y_emb, Wp, bp, pgen);
  // 8. big logits GEMM (WMMA, W2 streamed exactly once)
  wmma_gemm_nt<<<VV / 16, 256, 0, stream>>>(hid, W2, b2, logits, BB, VV, HH);
  // 9. softmax + final distribution assembly
  k_rowstats<<<BB, 256, 0, stream>>>(logits, mx, sm);
  k_final<<<dim3((VO + 255) / 256, BB), 256, 0, stream>>>(logits, mx, sm, pgen, fin);
  k_scatter_attn<<<(BB * SS + 255) / 256, 256, 0, stream>>>(src_oov, attn, pgen, fin);
  k_scatter_cls<<<(BB * CCN + 255) / 256, 256, 0, stream>>>(cls_oov, cls_pred, pgen, fin);
}